// TimeGraphConstructorMTGNN_18811956756533
// MI455X (gfx1250) — compile-verified
//
#include <hip/hip_runtime.h>
#include <math.h>

// Problem constants from the reference
#define BB  32
#define CIN 16
#define HH  64
#define SS  128

typedef __attribute__((ext_vector_type(2))) float v2f;
typedef __attribute__((ext_vector_type(8))) float v8f;

// ---------------------------------------------------------------------------
// Kernel 1: conv front-end -> hpre (B*S, 192)
// Recomputes the 1x1 conv rows (t1/t2) needed at the dilated tap positions,
// then the three dilated convs. SAME padding = zero rows at each conv input.
// One block of 64 threads per (b, s); thread c = output channel.
// ---------------------------------------------------------------------------
__global__ void front_kernel(const float* __restrict__ x,   // (B, CIN, S)
                             const float* __restrict__ w11, const float* __restrict__ b11,
                             const float* __restrict__ w12, const float* __restrict__ b12,
                             const float* __restrict__ w21, const float* __restrict__ b21,
                             const float* __restrict__ w22, const float* __restrict__ b22,
                             const float* __restrict__ w31, const float* __restrict__ b31,
                             float* __restrict__ hpre)       // (B*S, 192)
{
    const int bs = blockIdx.x;          // 0 .. B*S-1
    const int b  = bs >> 7;
    const int s  = bs & (SS - 1);
    const int c  = threadIdx.x;         // 0..63

    __shared__ float lx[11][CIN];       // x at offsets {-10,-5,-3,-2,-1,0,1,2,3,5,10}
    __shared__ float lt1[3][HH];        // conv11 output rows at s-3, s, s+3
    __shared__ float lt2[5][HH];        // conv21 output rows at s-10..s+10 step 5

    const int offs[11] = {-10, -5, -3, -2, -1, 0, 1, 2, 3, 5, 10};

    for (int t = c; t < 11 * CIN; t += 64) {
        const int r  = t / CIN;
        const int ci = t % CIN;
        const int sp = s + offs[r];
        lx[r][ci] = (sp >= 0 && sp < SS) ? x[b * CIN * SS + ci * SS + sp] : 0.0f;
    }
    __syncthreads();

    const int t1r[3] = {2, 5, 8};          // offsets -3, 0, +3
    const int t2r[5] = {0, 1, 5, 9, 10};   // offsets -10, -5, 0, +5, +10

#pragma unroll
    for (int r = 0; r < 3; ++r) {
        const int sp = s + offs[t1r[r]];
        float acc = 0.0f;
        if (sp >= 0 && sp < SS) {
            acc = b11[c];
#pragma unroll
            for (int ci = 0; ci < CIN; ++ci) acc += w11[c * CIN + ci] * lx[t1r[r]][ci];
        }
        lt1[r][c] = acc;                   // zero row outside => SAME padding for conv12
    }
#pragma unroll
    for (int r = 0; r < 5; ++r) {
        const int sp = s + offs[t2r[r]];
        float acc = 0.0f;
        if (sp >= 0 && sp < SS) {
            acc = b21[c];
#pragma unroll
            for (int ci = 0; ci < CIN; ++ci) acc += w21[c * CIN + ci] * lx[t2r[r]][ci];
        }
        lt2[r][c] = acc;
    }
    __syncthreads();

    // conv12: kernel 3, dilation 3 -> taps lt1[k] (k: -3,0,+3)
    float x1 = b12[c];
#pragma unroll
    for (int cp = 0; cp < HH; ++cp)
#pragma unroll
        for (int k = 0; k < 3; ++k)
            x1 += w12[c * HH * 3 + cp * 3 + k] * lt1[k][cp];

    // conv22: kernel 5, dilation 5 -> taps lt2[k]
    float x2 = b22[c];
#pragma unroll
    for (int cp = 0; cp < HH; ++cp)
#pragma unroll
        for (int k = 0; k < 5; ++k)
            x2 += w22[c * HH * 5 + cp * 5 + k] * lt2[k][cp];

    // conv31: kernel 5, dilation 1 on raw x -> lx rows 3..7 (offsets -2..+2)
    float x3 = b31[c];
#pragma unroll
    for (int ci = 0; ci < CIN; ++ci)
#pragma unroll
        for (int k = 0; k < 5; ++k)
            x3 += w31[c * CIN * 5 + ci * 5 + k] * lx[3 + k][ci];

    float* o = hpre + (size_t)bs * 192;
    o[c]        = x1;
    o[64 + c]   = x2;
    o[128 + c]  = x3;
}

// ---------------------------------------------------------------------------
// Kernel 2: fp32 WMMA GEMM  C(M,N) = A(M,K) @ W(N,K)^T [+bias][relu]
// One wave (32 lanes) per 16x16 output tile, V_WMMA_F32_16X16X4_F32.
// A fragment: lane holds A[m0+l][k+2*half .. +1]  (float2 load)
// B fragment: lane holds W[n0+l][k+2*half .. +1]  (float2 load)
// D layout  : VGPR r -> row m0 + r + 8*half, col n0 + l
// M, N multiples of 16; K multiple of 4; EXEC all-ones (no divergence).
// ---------------------------------------------------------------------------
__global__ void gemm_wmma_f32(const float* __restrict__ A, int lda,
                              const float* __restrict__ W, int ldw,
                              const float* __restrict__ bias,
                              float* __restrict__ C, int ldc,
                              int ntiles_n, int K, int do_relu)
{
    const int tile = blockIdx.x;
    const int tm = tile / ntiles_n;
    const int tn = tile % ntiles_n;
    const int lane = threadIdx.x & 31;
    const int half = lane >> 4;
    const int l    = lane & 15;

    const float* ap = A + (size_t)(tm * 16 + l) * lda + 2 * half;
    const float* wp = W + (size_t)(tn * 16 + l) * ldw + 2 * half;

    v8f acc = {};
    for (int k = 0; k < K; k += 4) {
        v2f a  = *(const v2f*)(ap + k);
        v2f bf = *(const v2f*)(wp + k);
        acc = __builtin_amdgcn_wmma_f32_16x16x4_f32(
            /*neg_a=*/false, a, /*neg_b=*/false, bf,
            /*c_mod=*/(short)0, acc, /*reuse_a=*/false, /*reuse_b=*/false);
    }

    const float bv = bias ? bias[tn * 16 + l] : 0.0f;
#pragma unroll
    for (int r = 0; r < 8; ++r) {
        const int row = tm * 16 + r + 8 * half;
        float v = acc[r] + bv;
        if (do_relu) v = fmaxf(v, 0.0f);
        C[(size_t)row * ldc + tn * 16 + l] = v;
    }
}

// ---------------------------------------------------------------------------
// Kernel 3: pair stage. For each (b, i, j):
//   v = relu(Hs[b][j] + Hr[b][i] + fc1_b); l0 = fc2_w0.v + fc2_b0; l1 = ...
//   out = (i < j) && (l0 + g0 >= l1 + g1) ? 1 : 0   (Gumbel-argmax, ties->0)
// 16x16 (i,j) tile per 256-thread block; Hs/Hr tiles staged in LDS (padded).
// ---------------------------------------------------------------------------
__global__ void pair_kernel(const float* __restrict__ Hs,  // (B*S, 64)
                            const float* __restrict__ Hr,  // (B*S, 64)
                            const float* __restrict__ fc1_b,
                            const float* __restrict__ fc2_w, // (2, 64)
                            const float* __restrict__ fc2_b, // (2)
                            const float* __restrict__ gum,   // (B, S*S, 2)
                            float* __restrict__ out)         // (B, S, S)
{
    const int jt = blockIdx.x;     // 0..7
    const int it = blockIdx.y;     // 0..7
    const int b  = blockIdx.z;     // 0..31
    const int tid = threadIdx.x;   // 0..255

    __shared__ float shs[16][HH + 1];
    __shared__ float shr[16][HH + 1];
    __shared__ float sb[HH];
    __shared__ float sw[2 * HH];

    const float* hsb = Hs + ((size_t)b * SS + jt * 16) * HH;
    const float* hrb = Hr + ((size_t)b * SS + it * 16) * HH;
    for (int t = tid; t < 16 * HH; t += 256) {
        shs[t >> 6][t & 63] = hsb[t];
        shr[t >> 6][t & 63] = hrb[t];
    }
    if (tid < HH)     sb[tid] = fc1_b[tid];
    if (tid < 2 * HH) sw[tid] = fc2_w[tid];
    __syncthreads();

    const int li = tid >> 4, lj = tid & 15;
    const int i = it * 16 + li;
    const int j = jt * 16 + lj;

    float l0 = fc2_b[0], l1 = fc2_b[1];
#pragma unroll 4
    for (int k = 0; k < HH; ++k) {
        float v = shs[lj][k] + shr[li][k] + sb[k];
        v = fmaxf(v, 0.0f);
        l0 += sw[k] * v;
        l1 += sw[HH + k] * v;
    }

    const int p = i * SS + j;
    const float* u = gum + (((size_t)b * SS * SS + p) << 1);
    const float g0 = -logf(-logf(u[0]));
    const float g1 = -logf(-logf(u[1]));

    const float r = ((i < j) && (l0 + g0 >= l1 + g1)) ? 1.0f : 0.0f;
    out[(size_t)b * SS * SS + p] = r;
}

// ---------------------------------------------------------------------------
// Host launcher
// ---------------------------------------------------------------------------
extern "C" void kernel_launch(void* const* d_in, const int* in_sizes, int n_in,
                              void* d_out, int out_size, void* d_ws, size_t ws_size,
                              hipStream_t stream) {
    const float* data  = (const float*)d_in[0];
    const float* w11   = (const float*)d_in[1];
    const float* b11   = (const float*)d_in[2];
    const float* w12   = (const float*)d_in[3];
    const float* b12   = (const float*)d_in[4];
    const float* w21   = (const float*)d_in[5];
    const float* b21   = (const float*)d_in[6];
    const float* w22   = (const float*)d_in[7];
    const float* b22   = (const float*)d_in[8];
    const float* w31   = (const float*)d_in[9];
    const float* b31   = (const float*)d_in[10];
    const float* fcw   = (const float*)d_in[11];  // (64, 192)
    const float* fcb   = (const float*)d_in[12];
    const float* fc1w  = (const float*)d_in[13];  // (64, 128)
    const float* fc1b  = (const float*)d_in[14];
    const float* fc2w  = (const float*)d_in[15];  // (2, 64)
    const float* fc2b  = (const float*)d_in[16];
    const float* gum   = (const float*)d_in[17];  // (32, 16384, 2)

    const int M = BB * SS;                 // 4096 flattened rows

    float* ws   = (float*)d_ws;
    float* hpre = ws;                      // 4096 * 192
    float* h    = hpre + (size_t)M * 192;  // 4096 * 64
    float* Hs   = h    + (size_t)M * HH;   // 4096 * 64
    float* Hr   = Hs   + (size_t)M * HH;   // 4096 * 64

    // 1) conv front-end -> hpre
    front_kernel<<<M, 64, 0, stream>>>(data, w11, b11, w12, b12, w21, b21,
                                       w22, b22, w31, b31, hpre);

    // 2) h = relu(hpre @ fc_final_w^T + fc_final_b)  : M=4096, N=64, K=192
    gemm_wmma_f32<<<(M / 16) * (HH / 16), 32, 0, stream>>>(
        hpre, 192, fcw, 192, fcb, h, HH, HH / 16, 192, 1);

    // 3) Hs = h @ fc1_w[:, :64]^T ; Hr = h @ fc1_w[:, 64:]^T  : K=64
    gemm_wmma_f32<<<(M / 16) * (HH / 16), 32, 0, stream>>>(
        h, HH, fc1w, 2 * HH, nullptr, Hs, HH, HH / 16, HH, 0);
    gemm_wmma_f32<<<(M / 16) * (HH / 16), 32, 0, stream>>>(
        h, HH, fc1w + HH, 2 * HH, nullptr, Hr, HH, HH / 16, HH, 0);

    // 4) pair stage + Gumbel argmax + strict upper-triangle mask
    pair_kernel<<<dim3(SS / 16, SS / 16, BB), 256, 0, stream>>>(
        Hs, Hr, fc1b, fc2w, fc2b, gum, (float*)d_out);
}